// SelfWiringLayer_31525059952785
// MI455X (gfx1250) — compile-verified
//
#include <hip/hip_runtime.h>

// SelfWiringLayer: out = segment_sum(x[src] * w, dst, 65536) + bias
//
// Memory-bound scatter/gather: 16.7M edges * 12B streamed = 201 MB @ 23.3 TB/s
// => ~8.6 us roofline. No dense matrix structure => WMMA inapplicable.
// CDNA5-specific design: 256 KB LDS output accumulator per workgroup (only
// possible with gfx1250's 320 KB/WGP LDS), ds_add_f32 privatized atomics,
// 4x fewer global atomics, non-temporal b128 edge streaming to protect the
// L2-resident x/out working set.

typedef int   v4i __attribute__((ext_vector_type(4)));
typedef float v4f __attribute__((ext_vector_type(4)));

#define NODES 65536   // SIZE in the reference
#define NWG   64      // 1 workgroup per WGP (256KB LDS each); 64 WGPs saturate HBM
#define TPB   1024    // 32 wave32 waves per WGP

__device__ __forceinline__ void lds_atomic_add(float* p, float v) {
  // relaxed, workgroup scope -> ds_add_f32 (no return)
  __hip_atomic_fetch_add(p, v, __ATOMIC_RELAXED, __HIP_MEMORY_SCOPE_WORKGROUP);
}

__device__ __forceinline__ void dev_atomic_add(float* p, float v) {
  // relaxed, agent scope -> global_atomic_add_f32 (fire-and-forget)
  __hip_atomic_fetch_add(p, v, __ATOMIC_RELAXED, __HIP_MEMORY_SCOPE_AGENT);
}

__global__ __launch_bounds__(TPB, 1)
void edge_scatter_kernel(const float* __restrict__ x,
                         const float* __restrict__ w,
                         const int*   __restrict__ src,
                         const int*   __restrict__ dst,
                         float*       __restrict__ out,
                         int nEdges) {
  // 256 KB private accumulator: covers the ENTIRE output vector.
  // Legal only on CDNA5 (320 KB LDS per WGP).
  __shared__ float acc[NODES];

  // Wide (ds_store_b128) zero-init: 16 DS ops per thread instead of 64.
  {
    v4f* acc4 = (v4f*)acc;
    const v4f z = {0.0f, 0.0f, 0.0f, 0.0f};
    for (int i = threadIdx.x; i < NODES / 4; i += TPB) acc4[i] = z;
  }
  __syncthreads();

  const v4i* __restrict__ src4 = (const v4i*)src;
  const v4i* __restrict__ dst4 = (const v4i*)dst;
  const v4f* __restrict__ w4   = (const v4f*)w;

  const int nQuads = nEdges >> 2;
  const int stride = (int)gridDim.x * TPB;          // quads per coalesced step
  int q = (int)blockIdx.x * TPB + (int)threadIdx.x;

  // Main loop: 4-way batched => 12 independent non-temporal b128 loads in
  // flight per thread (edge streams bypass L2 residency; x/out stay hot).
  for (; q + 3 * stride < nQuads; q += 4 * stride) {
    v4i s[4], d[4];
    v4f e[4];
#pragma unroll
    for (int k = 0; k < 4; ++k) {
      s[k] = __builtin_nontemporal_load(&src4[q + k * stride]);
      d[k] = __builtin_nontemporal_load(&dst4[q + k * stride]);
      e[k] = __builtin_nontemporal_load(&w4[q + k * stride]);
    }
    float xv[4][4];
#pragma unroll
    for (int k = 0; k < 4; ++k) {
      xv[k][0] = x[s[k].x];
      xv[k][1] = x[s[k].y];
      xv[k][2] = x[s[k].z];
      xv[k][3] = x[s[k].w];
    }
#pragma unroll
    for (int k = 0; k < 4; ++k) {
      lds_atomic_add(&acc[d[k].x], xv[k][0] * e[k].x);
      lds_atomic_add(&acc[d[k].y], xv[k][1] * e[k].y);
      lds_atomic_add(&acc[d[k].z], xv[k][2] * e[k].z);
      lds_atomic_add(&acc[d[k].w], xv[k][3] * e[k].w);
    }
  }

  // Quad tail
  for (; q < nQuads; q += stride) {
    v4i s = __builtin_nontemporal_load(&src4[q]);
    v4i d = __builtin_nontemporal_load(&dst4[q]);
    v4f e = __builtin_nontemporal_load(&w4[q]);
    lds_atomic_add(&acc[d.x], x[s.x] * e.x);
    lds_atomic_add(&acc[d.y], x[s.y] * e.y);
    lds_atomic_add(&acc[d.z], x[s.z] * e.z);
    lds_atomic_add(&acc[d.w], x[s.w] * e.w);
  }

  // Scalar tail (nEdges % 4)
  for (int t = (nQuads << 2) + (int)blockIdx.x * TPB + (int)threadIdx.x;
       t < nEdges; t += stride) {
    lds_atomic_add(&acc[dst[t]], x[src[t]] * w[t]);
  }

  __syncthreads();

  // Flush: wide ds_load_b128 reads, then 65536 device-scope adds per WG
  // (4.2M total vs 16.7M direct atomics = 4x cut). Avg 4 hits/bin => ~98%
  // of bins are non-zero; skip the empties anyway.
  {
    const v4f* acc4 = (const v4f*)acc;
    for (int i = threadIdx.x; i < NODES / 4; i += TPB) {
      v4f v = acc4[i];
      float* o = out + 4 * i;
      if (v.x != 0.0f) dev_atomic_add(o + 0, v.x);
      if (v.y != 0.0f) dev_atomic_add(o + 1, v.y);
      if (v.z != 0.0f) dev_atomic_add(o + 2, v.z);
      if (v.w != 0.0f) dev_atomic_add(o + 3, v.w);
    }
  }
}

__global__ void init_out_kernel(const float* __restrict__ bias,
                                float* __restrict__ out, int n) {
  int i = blockIdx.x * blockDim.x + threadIdx.x;
  if (i < n) out[i] = bias[i];
}

extern "C" void kernel_launch(void* const* d_in, const int* in_sizes, int n_in,
                              void* d_out, int out_size, void* d_ws, size_t ws_size,
                              hipStream_t stream) {
  const float* x    = (const float*)d_in[0];
  const float* w    = (const float*)d_in[1];
  const float* bias = (const float*)d_in[2];
  const int*   src  = (const int*)d_in[3];
  const int*   dst  = (const int*)d_in[4];
  float*       out  = (float*)d_out;

  const int nNodes = out_size;      // 65536
  const int nEdges = in_sizes[1];   // 16777216

  init_out_kernel<<<(nNodes + 255) / 256, 256, 0, stream>>>(bias, out, nNodes);
  edge_scatter_kernel<<<NWG, TPB, 0, stream>>>(x, w, src, dst, out, nEdges);
}